// Customize_GCN_67705864454449
// MI455X (gfx1250) — compile-verified
//
#include <hip/hip_runtime.h>
#include <hip/hip_bf16.h>
#include <math.h>

// ---------------------------------------------------------------------------
// GCN forward (2 layers, mean/add/min/max aggregation, log_softmax head)
// for MI455X (gfx1250, wave32).
//  - GEMMs: V_WMMA_F32_16X16X4_F32, one wave per 16x128 strip (8 accumulators)
//  - Weights staged to LDS via GLOBAL_LOAD_ASYNC_TO_LDS_B128 (ASYNCcnt)
//  - Edge scatter: global atomics (add + ordered-int min/max)
// ---------------------------------------------------------------------------

typedef float v2f __attribute__((ext_vector_type(2)));
typedef float v8f __attribute__((ext_vector_type(8)));

#define DH 128   // hidden dim
#define DCAT 512 // 4 * DH

// ---- float atomic min/max via order-preserving int mapping ----------------
__device__ __forceinline__ void atomicMaxF(float* addr, float val) {
    if (!(val < 0.0f)) atomicMax((int*)addr, __float_as_int(val));
    else               atomicMin((unsigned int*)addr, __float_as_uint(val));
}
__device__ __forceinline__ void atomicMinF(float* addr, float val) {
    if (!(val < 0.0f)) atomicMin((int*)addr, __float_as_int(val));
    else               atomicMax((unsigned int*)addr, __float_as_uint(val));
}

// ---- degree / normalization ------------------------------------------------
__global__ void deg_init_kernel(float* __restrict__ deg, int N) {
    int i = blockIdx.x * blockDim.x + threadIdx.x;
    if (i < N) deg[i] = 1.0f;  // self-loop contributes 1
}

__global__ void deg_accum_kernel(const int* __restrict__ col, float* __restrict__ deg, int E) {
    int e = blockIdx.x * blockDim.x + threadIdx.x;
    if (e < E) atomicAdd(&deg[col[e]], 1.0f);
}

__global__ void dinv_kernel(const float* __restrict__ deg, float* __restrict__ dinv, int N) {
    int i = blockIdx.x * blockDim.x + threadIdx.x;
    if (i < N) dinv[i] = rsqrtf(deg[i]);  // deg >= 1 always (self-loops)
}

// ---- aggregation buffer init: cat[n] = [mean | add | min | max], 512 floats
__global__ void cat_init_kernel(float* __restrict__ cat, int N) {
    int idx = blockIdx.x * blockDim.x + threadIdx.x;
    if (idx >= N * DH) return;
    int n = idx >> 7;        // / 128
    int f = idx & (DH - 1);  // % 128
    float* base = cat + (size_t)n * DCAT;
    base[0 * DH + f] = 0.0f;                       // mean (recomputed later)
    base[1 * DH + f] = 0.0f;                       // add
    base[2 * DH + f] = __int_as_float(0x7F800000); // +inf -> min
    base[3 * DH + f] = __int_as_float(0xFF800000); // -inf -> max
}

// ---- message scatter: one wave32 per edge, 4 features per lane -------------
__global__ void __launch_bounds__(256)
scatter_kernel(const float* __restrict__ h, const int* __restrict__ row,
               const int* __restrict__ col, const float* __restrict__ dinv,
               float* __restrict__ cat, int E, int N) {
    int lane = threadIdx.x & 31;
    int e = (blockIdx.x * blockDim.x + threadIdx.x) >> 5;
    int ET = E + N;
    if (e >= ET) return;
    int r, c;
    if (e < E) { r = row[e]; c = col[e]; }
    else       { r = c = e - E; }  // self-loop
    float norm = dinv[r] * dinv[c];
    const float* hr = h + (size_t)r * DH;
    float* base = cat + (size_t)c * DCAT;
#pragma unroll
    for (int j = 0; j < 4; ++j) {
        int f = lane + 32 * j;
        float v = norm * hr[f];
        atomicAdd(base + 1 * DH + f, v);
        atomicMinF(base + 2 * DH + f, v);
        atomicMaxF(base + 3 * DH + f, v);
    }
}

// ---- mean = add / deg ------------------------------------------------------
__global__ void finalize_mean_kernel(float* __restrict__ cat, const float* __restrict__ deg, int N) {
    int idx = blockIdx.x * blockDim.x + threadIdx.x;
    if (idx >= N * DH) return;
    int n = idx >> 7;
    int f = idx & (DH - 1);
    float* base = cat + (size_t)n * DCAT;
    base[f] = base[DH + f] / deg[n];  // deg >= 1
}

// ---------------------------------------------------------------------------
// Strip WMMA GEMM:  D[M,128] = A[M,K] @ W[128,K]^T (+bias) (+relu)
// Block = 256 threads = 8 waves; each wave computes a 16x128 strip (8 tiles,
// 8 x v8f accumulators). Weight matrix W (128 x K floats) is staged into LDS
// once per block with GLOBAL_LOAD_ASYNC_TO_LDS_B128, then B fragments are
// ds_load'ed; the A fragment is loaded once per k-step and reused across all
// 8 WMMAs (8x fewer A loads than tile-per-wave).
// f32 WMMA VGPR layouts per ISA 7.12.2:
//   A 16x4 : lane&15 = M, lane>>4 selects K-pair {k,k+1} vs {k+2,k+3}
//   B 4x16 : lane&15 = N, same K-pair split
//   C 16x16: VGPR r -> M = r + 8*(lane>>4), N = lane&15
// ---------------------------------------------------------------------------
template <int K, int RELU>
__global__ void __launch_bounds__(256)
wmma_gemm_strip_kernel(const float* __restrict__ A, const float* __restrict__ W,
                       const float* __restrict__ bias, float* __restrict__ D,
                       int M) {
    extern __shared__ float sW[];  // 128*K floats

    // ---- cooperative async stage of W into LDS (16B granules) ----
    {
        const int total = (DH * K) / 4;                 // number of b128 chunks
        unsigned ldsBase = (unsigned)(uintptr_t)sW;     // LDS offset = low 32 bits
        for (int i = threadIdx.x; i < total; i += 256) {
            unsigned byteOff = (unsigned)i * 16u;
            asm volatile(
                "global_load_async_to_lds_b128 %0, %1, %2"
                :
                : "v"(ldsBase + byteOff), "v"(byteOff), "s"(W)
                : "memory");
        }
        asm volatile("s_wait_asynccnt 0" ::: "memory");
    }
    __syncthreads();

    int lane = threadIdx.x & 31;
    int wv   = threadIdx.x >> 5;
    int mT   = blockIdx.x * 8 + wv;
    if (mT * 16 >= M) return;   // after the barrier: staging already done

    int half = lane >> 4;   // K-pair select
    int l    = lane & 15;   // M (A) / N (B) index

    const float* aRow  = A + (size_t)(mT * 16 + l) * K + 2 * half;
    const float* bBase = sW + 2 * half;

    v8f acc[8] = {};
    for (int k = 0; k < K; k += 4) {
        v2f a = *(const v2f*)(aRow + k);
#pragma unroll
        for (int j = 0; j < 8; ++j) {
            v2f b = *(const v2f*)(bBase + (size_t)(j * 16 + l) * K + k);
            acc[j] = __builtin_amdgcn_wmma_f32_16x16x4_f32(false, a, false, b,
                                                           (short)0, acc[j],
                                                           false, false);
        }
    }

#pragma unroll
    for (int j = 0; j < 8; ++j) {
        int colIdx = j * 16 + l;
        float bc = (bias != nullptr) ? bias[colIdx] : 0.0f;
#pragma unroll
        for (int r = 0; r < 8; ++r) {
            int m = mT * 16 + r + 8 * half;
            float v = acc[j][r] + bc;
            if (RELU) v = fmaxf(v, 0.0f);
            D[(size_t)m * DH + colIdx] = v;
        }
    }
}

// ---------------------------------------------------------------------------
// Tile WMMA GEMM for the 40-class head (Npad = 48, guarded columns).
// ---------------------------------------------------------------------------
__global__ void __launch_bounds__(128)
wmma_gemm_head_kernel(const float* __restrict__ A, const float* __restrict__ W,
                      const float* __restrict__ bias, float* __restrict__ D,
                      int M, int K, int Npad, int Nreal, int ldd) {
    int lane = threadIdx.x & 31;
    int wv   = threadIdx.x >> 5;
    int tile = blockIdx.x * (blockDim.x >> 5) + wv;
    int nTiles = Npad >> 4;
    int mT = tile / nTiles;
    int nT = tile - mT * nTiles;
    if (mT * 16 >= M) return;

    int half = lane >> 4;
    int l    = lane & 15;

    const float* aRow = A + (size_t)(mT * 16 + l) * K + 2 * half;
    int wcol = nT * 16 + l;
    bool wvalid = (wcol < Nreal);
    const float* bRow = W + (size_t)(wvalid ? wcol : 0) * K + 2 * half;
    float wsc = wvalid ? 1.0f : 0.0f;

    v8f c = {};
    for (int k = 0; k < K; k += 4) {
        v2f a = *(const v2f*)(aRow + k);
        v2f b = *(const v2f*)(bRow + k);
        b[0] *= wsc; b[1] *= wsc;
        c = __builtin_amdgcn_wmma_f32_16x16x4_f32(false, a, false, b,
                                                  (short)0, c, false, false);
    }

    float bc = (bias != nullptr && wvalid) ? bias[wcol] : 0.0f;
#pragma unroll
    for (int r = 0; r < 8; ++r) {
        int m = mT * 16 + r + 8 * half;
        float v = c[r] + bc;
        if (wvalid) D[(size_t)m * ldd + wcol] = v;
    }
}

// ---- log_softmax over 40 classes, one thread per node ----------------------
__global__ void logsoftmax_kernel(const float* __restrict__ logits,
                                  float* __restrict__ out, int N, int C) {
    int n = blockIdx.x * blockDim.x + threadIdx.x;
    if (n >= N) return;
    const float* l = logits + (size_t)n * C;
    float mx = -INFINITY;
    for (int c = 0; c < C; ++c) mx = fmaxf(mx, l[c]);
    float s = 0.0f;
    for (int c = 0; c < C; ++c) s += __expf(l[c] - mx);
    float lse = mx + __logf(s);
    float* o = out + (size_t)n * C;
    for (int c = 0; c < C; ++c) o[c] = l[c] - lse;
}

// ---------------------------------------------------------------------------
extern "C" void kernel_launch(void* const* d_in, const int* in_sizes, int n_in,
                              void* d_out, int out_size, void* d_ws, size_t ws_size,
                              hipStream_t stream) {
    const float* x    = (const float*)d_in[0];
    const int*   edge = (const int*)  d_in[1];
    const float* W0   = (const float*)d_in[2];
    const float* C0   = (const float*)d_in[3];
    const float* b0   = (const float*)d_in[4];
    const float* W1   = (const float*)d_in[5];
    const float* C1   = (const float*)d_in[6];
    const float* b1   = (const float*)d_in[7];
    const float* Wout = (const float*)d_in[8];
    const float* bout = (const float*)d_in[9];

    const int N = in_sizes[0] / DH;       // 40000
    const int E = in_sizes[1] / 2;        // 640000
    const int NCLS = in_sizes[9];         // 40
    const int* row = edge;
    const int* col = edge + E;

    float* ws = (float*)d_ws;
    float* deg    = ws;  ws += N;
    float* dinv   = ws;  ws += N;
    float* lin    = ws;  ws += (size_t)N * DH;
    float* act    = ws;  ws += (size_t)N * DH;
    float* cat    = ws;  ws += (size_t)N * DCAT;
    float* logits = ws;  ws += (size_t)N * NCLS;
    float* outp   = (float*)d_out;

    const int T = 256;
    dim3 blk(T);
    int gN    = (N + T - 1) / T;
    int gE    = (E + T - 1) / T;
    int gNF   = (N * DH + T - 1) / T;
    int gScat = ((E + N) * 32 + T - 1) / T;   // one wave32 per edge

    int mTiles = N / 16;                      // 2500
    int gStrip = (mTiles + 7) / 8;            // 8 strips per block
    size_t ldsW = (size_t)DH * DH   * sizeof(float);  //  64 KB
    size_t ldsC = (size_t)DH * DCAT * sizeof(float);  // 256 KB (fits 320 KB WGP LDS)

    int headTiles = mTiles * (48 / 16);
    int gHead = (headTiles + 3) / 4;

    // degree + normalization
    deg_init_kernel <<<gN, blk, 0, stream>>>(deg, N);
    deg_accum_kernel<<<gE, blk, 0, stream>>>(col, deg, E);
    dinv_kernel     <<<gN, blk, 0, stream>>>(deg, dinv, N);

    // ---------------- layer 0 ----------------
    wmma_gemm_strip_kernel<DH, 0><<<gStrip, 256, ldsW, stream>>>(x, W0, nullptr, lin, N);
    cat_init_kernel<<<gNF, blk, 0, stream>>>(cat, N);
    scatter_kernel <<<gScat, blk, 0, stream>>>(lin, row, col, dinv, cat, E, N);
    finalize_mean_kernel<<<gNF, blk, 0, stream>>>(cat, deg, N);
    wmma_gemm_strip_kernel<DCAT, 1><<<gStrip, 256, ldsC, stream>>>(cat, C0, b0, act, N);

    // ---------------- layer 1 ----------------
    wmma_gemm_strip_kernel<DH, 0><<<gStrip, 256, ldsW, stream>>>(act, W1, nullptr, lin, N);
    cat_init_kernel<<<gNF, blk, 0, stream>>>(cat, N);
    scatter_kernel <<<gScat, blk, 0, stream>>>(lin, row, col, dinv, cat, E, N);
    finalize_mean_kernel<<<gNF, blk, 0, stream>>>(cat, deg, N);
    wmma_gemm_strip_kernel<DCAT, 1><<<gStrip, 256, ldsC, stream>>>(cat, C1, b1, act, N);

    // ---------------- head + log_softmax ----------------
    wmma_gemm_head_kernel<<<gHead, 128, 0, stream>>>(act, Wout, bout, logits,
                                                     N, DH, 48, NCLS, NCLS);
    logsoftmax_kernel<<<gN, blk, 0, stream>>>(logits, outp, N, NCLS);
}